// TemporalHyperedge_14010183320108
// MI455X (gfx1250) — compile-verified
//
#include <hip/hip_runtime.h>
#include <math.h>

typedef __attribute__((ext_vector_type(2))) float v2f;
typedef __attribute__((ext_vector_type(8))) float v8f;

#define L2_LAMDA     0.001f
#define RECONS_LAMDA 0.2f
#define LB_TH        0.01f

// Problem sizes (fixed by the reference)
#define BATCH 32
#define NN    2048
#define DD    256

// Tiling
#define TM   128          // output rows (n) per workgroup
#define KT   32           // K chunk staged in LDS
#define KP   36           // padded K stride (16B-aligned rows, conflict-free frags)

// ---------------------------------------------------------------------------
// Kernel 1: per-row stats of masked W:  ws_rows[n] = l1max(n) + 0.001*l2(n)
// One wave (32 lanes) per row; 8 waves per block.
// ---------------------------------------------------------------------------
__global__ void rowstats_kernel(const float* __restrict__ inc,
                                float* __restrict__ ws_rows) {
    const int lane = threadIdx.x & 31;
    const int wid  = threadIdx.x >> 5;
    const int row  = blockIdx.x * 8 + wid;
    if (row >= NN) return;

    const float* r = inc + (size_t)row * NN;
    float l1 = 0.0f, l2 = 0.0f;
    for (int c = lane; c < NN; c += 32) {
        float w = r[c];
        w = (w > LB_TH) ? w : 0.0f;
        float aw = fabsf(w);
        l1 = fmaxf(l1, aw);
        l2 += w * w;
    }
    for (int m = 1; m < 32; m <<= 1) {
        l1 = fmaxf(l1, __shfl_xor(l1, m, 32));
        l2 += __shfl_xor(l2, m, 32);
    }
    if (lane == 0) ws_rows[row] = l1 + L2_LAMDA * sqrtf(l2);
}

// ---------------------------------------------------------------------------
// Kernel 2: fused  acc = W@pre - cur@r_proj  (per batch, 128-row tile),
// then rowwise sqrt(sum sq) summed over the tile -> partial per (b, tile).
// 512 threads = 16 waves in a 4(M) x 4(N) grid; each wave owns 2x4 = 8
// 16x16 f32 C-tiles (rows mw*32..+32, cols nw*64..+64).
// Double-buffered LDS with register-staged global loads for overlap.
// ---------------------------------------------------------------------------
__global__ __launch_bounds__(512)
void fused_recon_kernel(const float* __restrict__ cur,
                        const float* __restrict__ pre,
                        const float* __restrict__ r_proj,
                        const float* __restrict__ inc,
                        float* __restrict__ parts) {
    __shared__ float As[2][TM * KP];     // A tile: 128 x KT (row-major, padded)
    __shared__ float BsT[2][DD * KP];    // B tile transposed: 256(d) x KT(k)
    __shared__ float rowpart[4][TM];     // per N-quarter row sums of squares
    __shared__ float red[TM];

    const int tid  = threadIdx.x;
    const int lane = tid & 31;
    const int wid  = tid >> 5;           // 0..15
    const int mw   = wid >> 2;           // 0..3 -> rows [mw*32, mw*32+32)
    const int nw   = wid & 3;            // 0..3 -> cols [nw*64, nw*64+64)
    const int tile = blockIdx.x;         // 0..15 n-tile
    const int b    = blockIdx.y;         // 0..31 batch
    const int n0   = tile * TM;

    const int half = lane >> 4;          // 0/1: K- and M-half split per ISA layout
    const int l16  = lane & 15;

    v8f acc[2][4];
    #pragma unroll
    for (int mt = 0; mt < 2; ++mt)
        #pragma unroll
        for (int ct = 0; ct < 4; ++ct) acc[mt][ct] = (v8f)0.0f;

    const int NCH0 = NN / KT;            // 64 chunks: +W @ pre
    const int NCHT = NCH0 + DD / KT;     // +8 chunks: +(-cur) @ r_proj

    float4 aReg[2];                      // staged A granules (this thread)
    float4 bReg[4];                      // staged B granules (this thread)

    // ---- stage chunk c from global into registers ----
    auto loadRegs = [&](int c) {
        const int phase = (c >= NCH0) ? 1 : 0;
        const int k0    = phase ? (c - NCH0) * KT : c * KT;
        #pragma unroll
        for (int i = 0; i < 2; ++i) {            // A: 128x32 = 1024 float4
            int g  = tid + i * 512;
            int r  = g >> 3;
            int kq = (g & 7) << 2;
            float4 v;
            if (!phase) {
                v = *(const float4*)(inc + (size_t)(n0 + r) * NN + k0 + kq);
                v.x = (v.x > LB_TH) ? v.x : 0.0f;
                v.y = (v.y > LB_TH) ? v.y : 0.0f;
                v.z = (v.z > LB_TH) ? v.z : 0.0f;
                v.w = (v.w > LB_TH) ? v.w : 0.0f;
            } else {
                v = *(const float4*)(cur + ((size_t)b * NN + n0 + r) * DD + k0 + kq);
                v.x = -v.x; v.y = -v.y; v.z = -v.z; v.w = -v.w;
            }
            aReg[i] = v;
        }
        #pragma unroll
        for (int i = 0; i < 4; ++i) {            // B: 32x256 = 2048 float4
            int g  = tid + i * 512;
            int k  = g >> 6;
            int dq = (g & 63) << 2;
            bReg[i] = phase
                ? *(const float4*)(r_proj + (size_t)(k0 + k) * DD + dq)
                : *(const float4*)(pre + ((size_t)b * NN + k0 + k) * DD + dq);
        }
    };

    // ---- commit staged registers into LDS buffer p ----
    auto storeLds = [&](int p) {
        #pragma unroll
        for (int i = 0; i < 2; ++i) {
            int g  = tid + i * 512;
            int r  = g >> 3;
            int kq = (g & 7) << 2;
            *(float4*)&As[p][r * KP + kq] = aReg[i];
        }
        #pragma unroll
        for (int i = 0; i < 4; ++i) {
            int g  = tid + i * 512;
            int k  = g >> 6;
            int dq = (g & 63) << 2;
            BsT[p][(dq + 0) * KP + k] = bReg[i].x;
            BsT[p][(dq + 1) * KP + k] = bReg[i].y;
            BsT[p][(dq + 2) * KP + k] = bReg[i].z;
            BsT[p][(dq + 3) * KP + k] = bReg[i].w;
        }
    };

    // ---- 64 WMMAs over LDS buffer p ----
    auto compute = [&](int p) {
        #pragma unroll
        for (int ks = 0; ks < KT; ks += 4) {
            v2f a[2];
            #pragma unroll
            for (int mt = 0; mt < 2; ++mt)
                a[mt] = *(const v2f*)&As[p][(mw * 32 + mt * 16 + l16) * KP + ks + 2 * half];
            #pragma unroll
            for (int ct = 0; ct < 4; ++ct) {
                v2f bf = *(const v2f*)&BsT[p][(nw * 64 + ct * 16 + l16) * KP + ks + 2 * half];
                #pragma unroll
                for (int mt = 0; mt < 2; ++mt)
                    acc[mt][ct] = __builtin_amdgcn_wmma_f32_16x16x4_f32(
                        false, a[mt], false, bf, (short)0, acc[mt][ct], false, false);
            }
        }
    };

    // ---- software-pipelined main loop ----
    loadRegs(0);
    storeLds(0);
    __syncthreads();
    for (int c = 0; c < NCHT; ++c) {
        const int p = c & 1;
        if (c + 1 < NCHT) loadRegs(c + 1);   // global loads overlap WMMA below
        compute(p);
        if (c + 1 < NCHT) storeLds(p ^ 1);
        __syncthreads();
    }

    // ---- epilogue: per-row sum of squares over all D columns ----
    float s[2][8];
    #pragma unroll
    for (int mt = 0; mt < 2; ++mt)
        #pragma unroll
        for (int j = 0; j < 8; ++j) {
            float t = 0.0f;
            #pragma unroll
            for (int ct = 0; ct < 4; ++ct) t += acc[mt][ct][j] * acc[mt][ct][j];
            s[mt][j] = t;
        }
    #pragma unroll
    for (int m = 1; m < 16; m <<= 1)
        #pragma unroll
        for (int mt = 0; mt < 2; ++mt)
            #pragma unroll
            for (int j = 0; j < 8; ++j)
                s[mt][j] += __shfl_xor(s[mt][j], m, 32);
    if (l16 == 0) {
        #pragma unroll
        for (int mt = 0; mt < 2; ++mt)
            #pragma unroll
            for (int j = 0; j < 8; ++j)
                rowpart[nw][mw * 32 + mt * 16 + half * 8 + j] = s[mt][j];
    }
    __syncthreads();
    if (tid < TM)
        red[tid] = sqrtf(rowpart[0][tid] + rowpart[1][tid] +
                         rowpart[2][tid] + rowpart[3][tid]);
    __syncthreads();
    for (int stride = TM / 2; stride > 0; stride >>= 1) {
        if (tid < stride) red[tid] += red[tid + stride];
        __syncthreads();
    }
    if (tid == 0) parts[b * 16 + tile] = RECONS_LAMDA * red[0];
}

// ---------------------------------------------------------------------------
// Kernel 3: loss[b] = sum_n(l1+0.001*l2) + sum_tiles parts[b,:]
// ---------------------------------------------------------------------------
__global__ void finalize_kernel(const float* __restrict__ ws_rows,
                                const float* __restrict__ parts,
                                float* __restrict__ out) {
    __shared__ float sred[256];
    __shared__ float Cs;
    const int tid = threadIdx.x;
    float s = 0.0f;
    for (int i = tid; i < NN; i += 256) s += ws_rows[i];
    sred[tid] = s;
    __syncthreads();
    for (int stride = 128; stride > 0; stride >>= 1) {
        if (tid < stride) sred[tid] += sred[tid + stride];
        __syncthreads();
    }
    if (tid == 0) Cs = sred[0];
    __syncthreads();
    if (tid < BATCH) {
        float l = Cs;
        #pragma unroll
        for (int t = 0; t < 16; ++t) l += parts[tid * 16 + t];
        out[tid] = l;
    }
}

// ---------------------------------------------------------------------------
extern "C" void kernel_launch(void* const* d_in, const int* in_sizes, int n_in,
                              void* d_out, int out_size, void* d_ws, size_t ws_size,
                              hipStream_t stream) {
    const float* cur    = (const float*)d_in[0];
    const float* pre    = (const float*)d_in[1];
    const float* r_proj = (const float*)d_in[2];
    const float* inc    = (const float*)d_in[3];
    float* out = (float*)d_out;
    float* ws  = (float*)d_ws;

    float* ws_rows = ws;          // 2048 floats
    float* parts   = ws + NN;     // 32*16 = 512 floats

    rowstats_kernel<<<NN / 8, 256, 0, stream>>>(inc, ws_rows);

    dim3 grid(NN / TM, BATCH);    // (16, 32)
    fused_recon_kernel<<<grid, 512, 0, stream>>>(cur, pre, r_proj, inc, parts);

    finalize_kernel<<<1, 256, 0, stream>>>(ws_rows, parts, out);

    // Second output: incidence_m passed through, concatenated after loss (32 f32)
    hipMemcpyAsync(out + BATCH, inc, (size_t)NN * NN * sizeof(float),
                   hipMemcpyDeviceToDevice, stream);
}